// AttentionMLP_Autograd_54735063220781
// MI455X (gfx1250) — compile-verified
//
#include <hip/hip_runtime.h>

// MI455X / gfx1250, wave32. Two kernels:
//  1) qk_logits: batched QK^T via V_WMMA_F32_16X16X4_F32 (fp32-exact), masked store.
//  2) sigmoid_mean: per-channel sigmoid reduction over j (TRANS-bound), LDS-staged K.

typedef float v2f __attribute__((ext_vector_type(2)));
typedef float v8f __attribute__((ext_vector_type(8)));

#define B_ 2
#define I_ 1024
#define J_ 1024
#define C_ 64

// ---------------------------------------------------------------------------
// Kernel 1: attention_logits[b,i,j] = dot(Q[b,i,:], K[b,j,:]) * mask[b,i,j]
// One wave -> one 16x16 (i x j) tile, K=64 reduction = 16 x WMMA f32 16x16x4.
// Block = 256 threads = 8 waves -> 16(i) x 128(j) per block.
// ---------------------------------------------------------------------------
__global__ __launch_bounds__(256)
void qk_logits_kernel(const float* __restrict__ Q,
                      const float* __restrict__ K,
                      const float* __restrict__ mask,
                      float* __restrict__ out) {
  const int lane = threadIdx.x & 31;
  const int wave = threadIdx.x >> 5;
  const int hi   = lane >> 4;   // 0: lanes 0-15, 1: lanes 16-31
  const int lo   = lane & 15;

  const int b     = blockIdx.z;
  const int ibase = blockIdx.y * 16;
  const int jbase = blockIdx.x * 128 + wave * 16;

  // A-matrix (16x4 f32): lane lo = row M; VGPR0/1 hold K = 2*hi, 2*hi+1.
  const float* qrow = Q + ((size_t)b * I_ + (ibase + lo)) * C_;
  // B-matrix (4x16 f32): lane lo = col N; VGPR0/1 hold K = 2*hi, 2*hi+1.
  const float* krow = K + ((size_t)b * J_ + (jbase + lo)) * C_;

  v8f acc = {};
#pragma unroll
  for (int kb = 0; kb < C_; kb += 4) {
    v2f a  = *(const v2f*)(qrow + kb + 2 * hi);
    v2f bb = *(const v2f*)(krow + kb + 2 * hi);
    // 8 args: (neg_a, A, neg_b, B, c_mod, C, reuse_a, reuse_b)
    acc = __builtin_amdgcn_wmma_f32_16x16x4_f32(
        false, a, false, bb, (short)0, acc, false, false);
  }

  // C/D layout: VGPR r -> row (r + 8*hi), col lo.
#pragma unroll
  for (int r = 0; r < 8; ++r) {
    const int i = ibase + r + 8 * hi;
    const int j = jbase + lo;
    const size_t idx = ((size_t)b * I_ + i) * J_ + j;
    out[idx] = acc[r] * mask[idx];
  }
}

// ---------------------------------------------------------------------------
// Kernel 2: output[b,i,c] = sum_j sigmoid(Q[b,i,c]*K[b,j,c] + bias[c]) * m[b,i,j]
//                           / sum_j m[b,i,j]
// Block = 256 threads = 4 i-rows x 64 channels. K staged in LDS per 64-j chunk
// (lane c reads ldsK[jj*64+c]: conflict-free), mask tile broadcast from LDS.
// sigmoid = rcp(1 + exp2(-x * log2 e)): 2 TRANS ops/element.
// ---------------------------------------------------------------------------
__global__ __launch_bounds__(256)
void sigmoid_mean_kernel(const float* __restrict__ Q,
                         const float* __restrict__ K,
                         const float* __restrict__ bias,
                         const float* __restrict__ mask,
                         float* __restrict__ out) {
  __shared__ float ldsK[64 * 64];  // 16 KB: K[jb..jb+63][0..63]
  __shared__ float ldsM[4 * 64];   //  1 KB: mask[4 rows][64 j]

  const int tid = threadIdx.x;
  const int c   = tid & 63;
  const int ir  = tid >> 6;                 // 0..3
  const int b   = blockIdx.y;
  const int i   = blockIdx.x * 4 + ir;

  const float q  = Q[((size_t)b * I_ + i) * C_ + c];
  const float bc = bias[c];

  float acc = 0.0f;
  float msum = 0.0f;

  for (int jb = 0; jb < J_; jb += 64) {
    // Cooperative stage of 64x64 K chunk (4096 floats / 256 threads = 16 each).
#pragma unroll
    for (int t = 0; t < 16; ++t) {
      const int idx = tid + t * 256;
      const int jj = idx >> 6;
      const int cc = idx & 63;
      ldsK[idx] = K[((size_t)b * J_ + jb + jj) * C_ + cc];
    }
    // Stage 4x64 mask tile (one element per thread).
    {
      const int jj = tid & 63;
      const int r2 = tid >> 6;
      ldsM[tid] = mask[((size_t)b * I_ + blockIdx.x * 4 + r2) * J_ + jb + jj];
    }
    __syncthreads();

#pragma unroll 8
    for (int jj = 0; jj < 64; ++jj) {
      const float kv = ldsK[jj * 64 + c];
      const float m  = ldsM[ir * 64 + jj];
      const float x  = fmaf(q, kv, bc);
      const float e  = __builtin_amdgcn_exp2f(x * -1.44269504088896340736f);
      const float s  = __builtin_amdgcn_rcpf(1.0f + e);
      acc  = fmaf(m, s, acc);
      msum += m;
    }
    __syncthreads();
  }

  out[((size_t)b * I_ + i) * C_ + c] = acc / msum;
}

// ---------------------------------------------------------------------------
// Launch. Inputs (setup_inputs order): Q [B,I,C], K [B,J,C], bias [C],
// mask [B,I,J]. Outputs concatenated: output [B,I,C] then attention_logits
// [B,I,J].
// ---------------------------------------------------------------------------
extern "C" void kernel_launch(void* const* d_in, const int* in_sizes, int n_in,
                              void* d_out, int out_size, void* d_ws, size_t ws_size,
                              hipStream_t stream) {
  const float* Q    = (const float*)d_in[0];
  const float* K    = (const float*)d_in[1];
  const float* bias = (const float*)d_in[2];
  const float* mask = (const float*)d_in[3];

  float* out0 = (float*)d_out;                       // [B,I,C]
  float* out1 = out0 + (size_t)B_ * I_ * C_;         // [B,I,J]

  dim3 g1(J_ / 128, I_ / 16, B_);
  qk_logits_kernel<<<g1, dim3(256), 0, stream>>>(Q, K, mask, out1);

  dim3 g2(I_ / 4, B_);
  sigmoid_mean_kernel<<<g2, dim3(256), 0, stream>>>(Q, K, bias, mask, out0);
}